// HANLayer_90288802497382
// MI455X (gfx1250) — compile-verified
//
#include <hip/hip_runtime.h>

// ---------------- problem constants ----------------
constexpr int N_NODES = 100000;
constexpr int IN_F    = 128;
constexpr int OUT_F   = 64;
constexpr int P_MP    = 3;
constexpr int E_EDGES = 1600000;
constexpr int HID_F   = 32;

typedef __attribute__((ext_vector_type(2))) float v2f;
typedef __attribute__((ext_vector_type(8))) float v8f;

// ---------------- workspace layout (floats) ----------------
//  [0,                P*N)          deg/norm over src
//  [P*N,              2*P*N)        deg/norm over dst
//  [2*P*N,            2*P*N+N*OUT)  hw = h @ gc_w
//  [.. +P)                          wsum (per-metapath attention sums)
//  [.. +P)                          beta (softmax weights)
constexpr size_t OFF_NS   = 0;
constexpr size_t OFF_ND   = (size_t)P_MP * N_NODES;
constexpr size_t OFF_HW   = 2ull * P_MP * N_NODES;
constexpr size_t OFF_WSUM = OFF_HW + (size_t)N_NODES * OUT_F;
constexpr size_t OFF_BETA = OFF_WSUM + P_MP;

// ---------------- kernels ----------------
__global__ void zero_kernel(float* __restrict__ p, size_t n) {
  size_t i = (size_t)blockIdx.x * blockDim.x + threadIdx.x;
  size_t stride = (size_t)gridDim.x * blockDim.x;
  for (; i < n; i += stride) p[i] = 0.0f;
}

// degree accumulation over all P graphs at once
__global__ void degree_kernel(const int* __restrict__ src,
                              const int* __restrict__ dst,
                              float* __restrict__ deg_src,
                              float* __restrict__ deg_dst) {
  long i = (long)blockIdx.x * blockDim.x + threadIdx.x;
  if (i >= (long)P_MP * E_EDGES) return;
  int p = (int)(i / E_EDGES);
  atomicAdd(&deg_src[(size_t)p * N_NODES + src[i]], 1.0f);
  atomicAdd(&deg_dst[(size_t)p * N_NODES + dst[i]], 1.0f);
}

// deg -> rsqrt(max(deg,1)) in place (covers both src and dst arrays)
__global__ void norm_kernel(float* __restrict__ deg, size_t n) {
  size_t i = (size_t)blockIdx.x * blockDim.x + threadIdx.x;
  if (i >= n) return;
  deg[i] = __frsqrt_rn(fmaxf(deg[i], 1.0f));
}

// hw[N,64] = h[N,128] @ gc_w[128,64] via V_WMMA_F32_16X16X4_F32.
// 256 threads = 8 waves; block tile = 32 rows x 64 cols (2x4 wave tiles).
// N % 32 == 0 so no remainder handling; EXEC stays all-ones (WMMA requirement).
__global__ __launch_bounds__(256)
void gemm_wmma_kernel(const float* __restrict__ h,
                      const float* __restrict__ w,
                      float* __restrict__ hw) {
  const int wave = threadIdx.x >> 5;        // 0..7
  const int lane = threadIdx.x & 31;
  const int row0 = (blockIdx.x * 2 + (wave >> 2)) * 16;  // 16-row tile base
  const int col0 = (wave & 3) * 16;                      // 16-col tile base

  const int m     = lane & 15;          // M (for A) / N (for B,D) within tile
  const int khalf = (lane >> 4) * 2;    // lanes 16..31 carry K+2 / K+3

  const float* __restrict__ arow = h + (size_t)(row0 + m) * IN_F;

  v8f c = {};
  #pragma unroll
  for (int k = 0; k < IN_F; k += 4) {
    v2f a, b;
    // A 16x4 f32 fragment: VGPR0 = K{0|2}, VGPR1 = K{1|3} per documented layout
    a.x = arow[k + khalf];
    a.y = arow[k + khalf + 1];
    // B 4x16 f32 fragment: row K striped across lanes, mirrored K split
    b.x = w[(size_t)(k + khalf)     * OUT_F + col0 + m];
    b.y = w[(size_t)(k + khalf + 1) * OUT_F + col0 + m];
    c = __builtin_amdgcn_wmma_f32_16x16x4_f32(
        /*neg_a=*/false, a, /*neg_b=*/false, b,
        /*c_mod=*/(short)0, c, /*reuse_a=*/false, /*reuse_b=*/false);
  }

  // D layout: VGPR r -> M = r (+8 for lanes 16..31), N = lane&15
  const int nout  = col0 + m;
  const int rbase = row0 + ((lane >> 4) << 3);
  #pragma unroll
  for (int r = 0; r < 8; ++r)
    hw[(size_t)(rbase + r) * OUT_F + nout] = c[r];
}

// scatter-add: one wave per edge, 2 features per lane (contiguous 256B runs)
// accumulates norm_src[src]*hw[src] into z[dst, p, :] (z lives inside d_out)
__global__ __launch_bounds__(256)
void scatter_kernel(const int* __restrict__ src,
                    const int* __restrict__ dst,
                    const float* __restrict__ norm_src,
                    const float* __restrict__ hw,
                    float* __restrict__ z) {
  long t = (long)blockIdx.x * blockDim.x + threadIdx.x;
  const long total = (long)P_MP * E_EDGES * 32;
  if (t >= total) return;
  const int  lane = (int)(t & 31);
  const long e    = t >> 5;                  // edge index over all graphs
  const int  p    = (int)(e / E_EDGES);
  const int  s    = src[e];
  const int  d    = dst[e];
  const float scale = norm_src[(size_t)p * N_NODES + s];
  const float* __restrict__ f = hw + (size_t)s * OUT_F + lane * 2;
  float* __restrict__ zp = z + (size_t)d * (P_MP * OUT_F) + p * OUT_F + lane * 2;
  atomicAdd(&zp[0], scale * f[0]);
  atomicAdd(&zp[1], scale * f[1]);
}

// z = relu(z * norm_dst + gc_b), in place over [N, P, OUT]
__global__ void finalize_z_kernel(float* __restrict__ z,
                                  const float* __restrict__ norm_dst,
                                  const float* __restrict__ gc_b) {
  size_t i = (size_t)blockIdx.x * blockDim.x + threadIdx.x;
  if (i >= (size_t)N_NODES * P_MP * OUT_F) return;
  const int n   = (int)(i / (P_MP * OUT_F));
  const int rem = (int)(i % (P_MP * OUT_F));
  const int p   = rem / OUT_F;
  const int o   = rem % OUT_F;
  z[i] = fmaxf(z[i] * norm_dst[(size_t)p * N_NODES + n] + gc_b[o], 0.0f);
}

// per-node attention MLP, reduce sum_n w[n,p] into wsum[p]
__global__ __launch_bounds__(256)
void attn_reduce_kernel(const float* __restrict__ z,
                        const float* __restrict__ sa_w1,
                        const float* __restrict__ sa_b1,
                        const float* __restrict__ sa_w2,
                        float* __restrict__ wsum) {
  __shared__ float red[P_MP];
  if (threadIdx.x < P_MP) red[threadIdx.x] = 0.0f;
  __syncthreads();
  const int n = blockIdx.x * blockDim.x + threadIdx.x;
  if (n < N_NODES) {
    for (int p = 0; p < P_MP; ++p) {
      const float* __restrict__ zp = z + (size_t)n * (P_MP * OUT_F) + p * OUT_F;
      float hid[HID_F];
      #pragma unroll
      for (int t = 0; t < HID_F; ++t) hid[t] = sa_b1[t];
      for (int o = 0; o < OUT_F; ++o) {
        const float zv = zp[o];
        #pragma unroll
        for (int t = 0; t < HID_F; ++t) hid[t] += zv * sa_w1[o * HID_F + t];
      }
      float wacc = 0.0f;
      #pragma unroll
      for (int t = 0; t < HID_F; ++t) wacc += tanhf(hid[t]) * sa_w2[t];
      atomicAdd(&red[p], wacc);
    }
  }
  __syncthreads();
  if (threadIdx.x < P_MP) atomicAdd(&wsum[threadIdx.x], red[threadIdx.x]);
}

// beta = softmax(wsum / N) over the P meta-paths
__global__ void beta_kernel(const float* __restrict__ wsum, float* __restrict__ beta) {
  float w[P_MP], m = -3.4e38f;
  for (int p = 0; p < P_MP; ++p) { w[p] = wsum[p] / (float)N_NODES; m = fmaxf(m, w[p]); }
  float s = 0.0f, e[P_MP];
  for (int p = 0; p < P_MP; ++p) { e[p] = expf(w[p] - m); s += e[p]; }
  for (int p = 0; p < P_MP; ++p) beta[p] = e[p] / s;
}

// out[n,o] = sum_p beta[p] * z[n,p,o]
__global__ void output_kernel(const float* __restrict__ z,
                              const float* __restrict__ beta,
                              float* __restrict__ out) {
  size_t i = (size_t)blockIdx.x * blockDim.x + threadIdx.x;
  if (i >= (size_t)N_NODES * OUT_F) return;
  const int n = (int)(i / OUT_F);
  const int o = (int)(i % OUT_F);
  const float* __restrict__ zn = z + (size_t)n * (P_MP * OUT_F) + o;
  out[i] = beta[0] * zn[0] + beta[1] * zn[OUT_F] + beta[2] * zn[2 * OUT_F];
}

// ---------------- launcher ----------------
extern "C" void kernel_launch(void* const* d_in, const int* in_sizes, int n_in,
                              void* d_out, int out_size, void* d_ws, size_t ws_size,
                              hipStream_t stream) {
  const float* h     = (const float*)d_in[0];
  const float* gc_w  = (const float*)d_in[1];
  const float* gc_b  = (const float*)d_in[2];
  const float* sa_w1 = (const float*)d_in[3];
  const float* sa_b1 = (const float*)d_in[4];
  const float* sa_w2 = (const float*)d_in[5];
  const int*   esrc  = (const int*)d_in[6];
  const int*   edst  = (const int*)d_in[7];

  float* ws   = (float*)d_ws;
  float* ns   = ws + OFF_NS;    // norm_src  [P,N]
  float* nd   = ws + OFF_ND;    // norm_dst  [P,N]
  float* hw   = ws + OFF_HW;    // h @ gc_w  [N,OUT]
  float* wsum = ws + OFF_WSUM;  // [P]
  float* beta = ws + OFF_BETA;  // [P]

  float* out_buf = (float*)d_out;                             // [N,OUT]
  float* z_buf   = (float*)d_out + (size_t)N_NODES * OUT_F;   // [N,P,OUT]

  const int TB = 256;

  // 1) zero accumulators (degrees, wsum, z scatter target)
  zero_kernel<<<2048, TB, 0, stream>>>(ns, 2ull * P_MP * N_NODES);
  zero_kernel<<<1, TB, 0, stream>>>(wsum, (size_t)P_MP);
  zero_kernel<<<8192, TB, 0, stream>>>(z_buf, (size_t)N_NODES * P_MP * OUT_F);

  // 2) degrees for all P graphs
  {
    long total = (long)P_MP * E_EDGES;
    degree_kernel<<<(int)((total + TB - 1) / TB), TB, 0, stream>>>(esrc, edst, ns, nd);
  }

  // 3) deg -> rsqrt(max(deg,1))
  {
    size_t total = 2ull * P_MP * N_NODES;
    norm_kernel<<<(int)((total + TB - 1) / TB), TB, 0, stream>>>(ns, total);
  }

  // 4) hw = h @ gc_w (fp32 WMMA)
  gemm_wmma_kernel<<<N_NODES / 32, TB, 0, stream>>>(h, gc_w, hw);

  // 5) scatter-add norm_src[src]*hw[src] into z[dst,p,:] — one wave per edge
  {
    long total = (long)P_MP * E_EDGES * 32;
    scatter_kernel<<<(int)((total + TB - 1) / TB), TB, 0, stream>>>(esrc, edst, ns, hw, z_buf);
  }

  // 6) z = relu(z * norm_dst + gc_b)
  {
    size_t total = (size_t)N_NODES * P_MP * OUT_F;
    finalize_z_kernel<<<(int)((total + TB - 1) / TB), TB, 0, stream>>>(z_buf, nd, gc_b);
  }

  // 7) semantic attention reduction
  attn_reduce_kernel<<<(N_NODES + TB - 1) / TB, TB, 0, stream>>>(z_buf, sa_w1, sa_b1, sa_w2, wsum);

  // 8) beta = softmax(mean(w))
  beta_kernel<<<1, 1, 0, stream>>>(wsum, beta);

  // 9) out = sum_p beta[p] * z[:,p,:]
  {
    size_t total = (size_t)N_NODES * OUT_F;
    output_kernel<<<(int)((total + TB - 1) / TB), TB, 0, stream>>>(z_buf, beta, out_buf);
  }
}